// R2KD_71829033058968
// MI455X (gfx1250) — compile-verified
//
#include <hip/hip_runtime.h>
#include <math.h>

typedef __attribute__((ext_vector_type(2))) float v2f;
typedef __attribute__((ext_vector_type(8))) float v8f;

#define NN 512
#define CC 1000
#define MARGIN 0.1f
#define ATILES 32              // 512 / 16
#define NCHUNK ((NN * NN) / 16)  // 16384 chunks of 16 d-values

// ---------------------------------------------------------------------------
// Kernel 1: per-row argmax of t_logits -> ins_lv; extract diag(dist); zero acc
// One wave (32 lanes) per row.
// ---------------------------------------------------------------------------
__global__ void rk_prep(const int* __restrict__ gt,
                        const float* __restrict__ t_logits,
                        const float* __restrict__ dist,
                        float* __restrict__ acc,      // [32] per-a-tile sums
                        float* __restrict__ a_diag,   // [512]
                        float* __restrict__ ins_lv) { // [512]
    const int tid  = blockIdx.x * blockDim.x + threadIdx.x;
    const int lane = threadIdx.x & 31;
    const int row  = tid >> 5;

    if (blockIdx.x == 0 && threadIdx.x < ATILES) acc[threadIdx.x] = 0.0f;

    if (row < NN) {
        const float* tl = t_logits + (long)row * CC;
        float best  = -INFINITY;
        int   besti = 0x7fffffff;
        for (int c = lane; c < CC; c += 32) {
            float v = tl[c];
            if (v > best) { best = v; besti = c; }
        }
        // butterfly reduce: max value, ties -> smallest index (jnp.argmax)
        #pragma unroll
        for (int off = 16; off > 0; off >>= 1) {
            float ov = __shfl_xor(best, off, 32);
            int   oi = __shfl_xor(besti, off, 32);
            if (ov > best || (ov == best && oi < besti)) { best = ov; besti = oi; }
        }
        if (lane == 0) {
            ins_lv[row] = (besti == gt[row]) ? 2.0f : 1.0f;
            a_diag[row] = dist[(long)row * NN + row];
        }
    }
}

// ---------------------------------------------------------------------------
// Kernel 2: S = sum_i sum_{jk} relu(a_i + M - d_jk) over ALL (j,k).
// Per wave: fixed 16 a-values (a-tile = blockIdx.x), stream 16 d-values/iter.
// T[m,n] = a_m + M - d_n via V_WMMA_F32_16X16X4_F32 (rank-2 trick):
//   A[m,0]=a_m+M, A[m,1]=1 (K=2,3 zero);  B[0,n]=1, B[1,n]=-d_n.
// ---------------------------------------------------------------------------
__global__ void rk_main(const float* __restrict__ dist,
                        const float* __restrict__ a_diag,
                        float* __restrict__ acc) {
    const int lane   = threadIdx.x & 31;
    const int waveib = threadIdx.x >> 5;      // 0..7 (8 waves / block)
    const int atile  = blockIdx.x;            // 0..31

    // A-matrix (loop invariant): lanes 0-15 hold M=0-15 {K0,K1}; lanes 16-31 -> K2,K3 = 0
    const bool lo = (lane < 16);
    v2f A;
    A.x = lo ? (a_diag[atile * 16 + (lane & 15)] + MARGIN) : 0.0f;  // K=0 col: a_m + M
    A.y = lo ? 1.0f : 0.0f;                                          // K=1 col: 1

    v8f accv = {};  // 16x16 relu-sum accumulator (8 VGPRs/lane)

    const int start  = blockIdx.y * 8 + waveib;   // wave-uniform
    const int stride = gridDim.y * 8;

    for (int c = start; c < NCHUNK; c += stride) {
        const float dval = dist[c * 16 + (lane & 15)];
        v2f B;
        B.x = lo ? 1.0f  : 0.0f;   // K=0 row of B: ones
        B.y = lo ? -dval : 0.0f;   // K=1 row of B: -d_n
        v8f zero = {};
        // D[m,n] = a_m + MARGIN - d_n
        v8f T = __builtin_amdgcn_wmma_f32_16x16x4_f32(
            /*neg_a=*/false, A, /*neg_b=*/false, B,
            /*c_mod=*/(short)0, zero, /*reuse_a=*/false, /*reuse_b=*/false);
        #pragma unroll
        for (int r = 0; r < 8; ++r) accv[r] += fmaxf(T[r], 0.0f);
    }

    float s = 0.0f;
    #pragma unroll
    for (int r = 0; r < 8; ++r) s += accv[r];
    #pragma unroll
    for (int off = 16; off > 0; off >>= 1) s += __shfl_xor(s, off, 32);

    if (lane == 0) atomicAdd(&acc[atile], s);   // global_atomic_add_f32
}

// ---------------------------------------------------------------------------
// Kernel 3 (single block): diagonal correction + final mean.
//   corr = sum_ij [ relu(y_ij*(d_ii - d_jj) + M) - relu(d_ii - d_jj + M) ]
//   out  = (sum(acc) + corr) / N^3
// ---------------------------------------------------------------------------
__global__ void rk_final(const float* __restrict__ a_diag,
                         const float* __restrict__ ins_lv,
                         const float* __restrict__ acc,
                         float* __restrict__ out) {
    __shared__ float red[256];
    float s = 0.0f;
    for (int idx = threadIdx.x; idx < NN * NN; idx += blockDim.x) {
        const int i = idx >> 9;
        const int j = idx & (NN - 1);
        const float di = a_diag[i], dj = a_diag[j];
        const float li = ins_lv[i], lj = ins_lv[j];
        float y = li - lj;
        y = (y > 0.0f) ? 1.0f : y;
        const float diagloss = fmaxf(0.0f, y * (di - dj) + MARGIN);
        const float wrong    = fmaxf(0.0f, (di - dj) + MARGIN);
        s += diagloss - wrong;
    }
    red[threadIdx.x] = s;
    __syncthreads();
    for (int off = 128; off > 0; off >>= 1) {
        if (threadIdx.x < off) red[threadIdx.x] += red[threadIdx.x + off];
        __syncthreads();
    }
    if (threadIdx.x == 0) {
        double total = (double)red[0];
        for (int t = 0; t < ATILES; ++t) total += (double)acc[t];
        out[0] = (float)(total / ((double)NN * (double)NN * (double)NN));
    }
}

// ---------------------------------------------------------------------------
extern "C" void kernel_launch(void* const* d_in, const int* in_sizes, int n_in,
                              void* d_out, int out_size, void* d_ws, size_t ws_size,
                              hipStream_t stream) {
    const int*   gt       = (const int*)d_in[0];
    const float* t_logits = (const float*)d_in[1];
    // d_in[2] (s_logits) is unused by the reference computation.
    const float* dist     = (const float*)d_in[3];
    float* out = (float*)d_out;

    float* w      = (float*)d_ws;
    float* acc    = w;               // 32 floats (per-a-tile accumulators)
    float* a_diag = w + 32;          // 512 floats
    float* ins_lv = w + 32 + NN;     // 512 floats

    // 512 rows, one wave per row -> 16384 threads
    rk_prep<<<64, 256, 0, stream>>>(gt, t_logits, dist, acc, a_diag, ins_lv);
    // 32 a-tiles x 32 slices, 8 waves/block -> 8192 waves, 64 WMMA iters each
    rk_main<<<dim3(ATILES, 32), 256, 0, stream>>>(dist, a_diag, acc);
    rk_final<<<1, 256, 0, stream>>>(a_diag, ins_lv, acc, out);
}